// JpegLayer_74105365725253
// MI455X (gfx1250) — compile-verified
//
#include <hip/hip_runtime.h>
#include <hip/hip_bf16.h>
#include <math.h>

// ---------------------------------------------------------------------------
// JPEG round-trip layer for MI455X (gfx1250), fp32 end-to-end.
// One wave32 handles one 16x16 pixel tile (all 3 channels).
// 8x8 DCT/IDCT done as 16x16 WMMA chains with a block-diagonal DCT matrix,
// using V_WMMA_F32_16X16X4_F32 (K=16 via 4 chained K=4 slices).
// Bandwidth-bound problem (~201 MB round trip -> ~8.6us floor @ 23.3 TB/s):
// pixel traffic is non-temporal; transposes use per-wave-private LDS with
// wave-local ordering (s_wait_dscnt) instead of workgroup barriers.
// ---------------------------------------------------------------------------

typedef float v2f __attribute__((ext_vector_type(2)));
typedef float v8f __attribute__((ext_vector_type(8)));

#define IMG_B 32
#define IMG_H 512
#define IMG_W 512
#define HW    (IMG_H * IMG_W)
#define TILES_PER_IMG ((IMG_H / 16) * (IMG_W / 16))   // 1024
#define TOTAL_TILES   (IMG_B * TILES_PER_IMG)          // 32768
#define WAVES_PER_BLOCK 8
#define BLOCKS (TOTAL_TILES / WAVES_PER_BLOCK)         // 4096
#define LDS_PITCH 17                                   // 16+1 pad: conflict-free transpose

__device__ __forceinline__ v8f wmma4(float a0, float a1, float b0, float b1, v8f acc) {
    v2f a = {a0, a1};
    v2f b = {b0, b1};
    // D(16x16,f32) = A(16x4,f32) x B(4x16,f32) + C
    return __builtin_amdgcn_wmma_f32_16x16x4_f32(
        /*neg_a=*/false, a, /*neg_b=*/false, b,
        /*c_mod=*/(short)0, acc, /*reuse_a=*/false, /*reuse_b=*/false);
}

// JPEG quality-scaled quantizer (QUALITY = 50): round((round(q*255)*50+50)/100)
// clipped to [1,255]; clamp of an integer-valued rintf result stays integral,
// so the reference's outer round() is a no-op.
__device__ __forceinline__ float quant_q(float qraw) {
    float t = rintf(fmaf(rintf(qraw * 255.0f), 0.5f, 0.5f));
    return fminf(fmaxf(t, 1.0f), 255.0f);
}

// C/D layout (reg r -> row g*8+r, col = lane&15) -> B layout of T^T
// (reg kk*2+j -> row 4kk+2g+j of T^T = column of T), via padded LDS.
// The LDS region is private to this wave and same-wave LDS ops execute in
// order (ISA 7.3), so wave-local s_wait_dscnt + compiler fences suffice —
// no workgroup barrier, waves stay fully independent.
__device__ __forceinline__ void wave_transpose(float* ldsw, int g, int c,
                                               const v8f& cd, float tb[8]) {
#pragma unroll
    for (int r = 0; r < 8; ++r)
        ldsw[(g * 8 + r) * LDS_PITCH + c] = cd[r];
    asm volatile("s_wait_dscnt 0x0" ::: "memory");
#pragma unroll
    for (int kk = 0; kk < 4; ++kk)
#pragma unroll
        for (int j = 0; j < 2; ++j)
            tb[kk * 2 + j] = ldsw[c * LDS_PITCH + (4 * kk + 2 * g + j)];
    asm volatile("" ::: "memory");
}

// Full per-channel pipeline: s[] holds the 16x16 sample tile in B layout.
// Computes Z = Mt * quantdq( M * X * Mt ) * M and returns Z in C/D layout.
__device__ __forceinline__ void dct_pipeline(const float s[8], const float aM[8],
                                             const float aMt[8], const float* qch,
                                             int g, int c, float* ldsw, float zout[8]) {
    // chain1: P = M * X            (X fed as B operand directly)
    v8f acc1 = {0.f, 0.f, 0.f, 0.f, 0.f, 0.f, 0.f, 0.f};
#pragma unroll
    for (int kk = 0; kk < 4; ++kk)
        acc1 = wmma4(aM[2 * kk], aM[2 * kk + 1], s[2 * kk], s[2 * kk + 1], acc1);

    float tb[8];
    wave_transpose(ldsw, g, c, acc1, tb);          // tb = B layout of P^T

    // chain2: Q = M * P^T = (M X Mt)^T = Y^T
    v8f acc2 = {0.f, 0.f, 0.f, 0.f, 0.f, 0.f, 0.f, 0.f};
#pragma unroll
    for (int kk = 0; kk < 4; ++kk)
        acc2 = wmma4(aM[2 * kk], aM[2 * kk + 1], tb[2 * kk], tb[2 * kk + 1], acc2);

    // quant/dequant (elementwise; Q holds Y transposed: u = c&7, v = r).
    // v_rcp_f32 instead of IEEE division: q is an integer in [1,255] and the
    // result feeds rintf, so ~1ulp rcp error is immaterial.
    float cd2[8];
#pragma unroll
    for (int r = 0; r < 8; ++r) {
        float qv   = quant_q(qch[(c & 7) * 8 + r]);
        float comp = rintf(acc2[r] * __builtin_amdgcn_rcpf(qv));
        cd2[r]     = rintf(comp * qv);
    }

    // cheap remap: C/D of D2 -> B layout of D2 (same matrix, row re-index)
    float bb[8];
#pragma unroll
    for (int kk = 0; kk < 4; ++kk)
#pragma unroll
        for (int j = 0; j < 2; ++j) {
            const int R0 = 4 * kk + j;      // row needed by lanes with g==0
            const int R1 = R0 + 2;          // row needed by lanes with g==1
            float v0 = (R0 < 8) ? cd2[R0] : __shfl_xor(cd2[R0 - 8], 16, 32);
            float v1 = (R1 >= 8) ? cd2[R1 - 8] : __shfl_xor(cd2[R1], 16, 32);
            bb[kk * 2 + j] = (g == 0) ? v0 : v1;
        }

    // chain3: P3 = Mt * D2 = Mt * Y~^T
    v8f acc3 = {0.f, 0.f, 0.f, 0.f, 0.f, 0.f, 0.f, 0.f};
#pragma unroll
    for (int kk = 0; kk < 4; ++kk)
        acc3 = wmma4(aMt[2 * kk], aMt[2 * kk + 1], bb[2 * kk], bb[2 * kk + 1], acc3);

    wave_transpose(ldsw, g, c, acc3, tb);          // tb = B layout of P3^T

    // chain4: Z = Mt * P3^T = Mt * Y~ * M   (IDCT, spatial C/D layout)
    v8f acc4 = {0.f, 0.f, 0.f, 0.f, 0.f, 0.f, 0.f, 0.f};
#pragma unroll
    for (int kk = 0; kk < 4; ++kk)
        acc4 = wmma4(aMt[2 * kk], aMt[2 * kk + 1], tb[2 * kk], tb[2 * kk + 1], acc4);

#pragma unroll
    for (int r = 0; r < 8; ++r) zout[r] = acc4[r];
}

__global__ __launch_bounds__(WAVES_PER_BLOCK * 32) void jpeg_layer_kernel(
    const float* __restrict__ in, const float* __restrict__ qtab,
    const float* __restrict__ dct, float* __restrict__ out) {
    __shared__ float lds[WAVES_PER_BLOCK][16 * LDS_PITCH];

    const int lane = threadIdx.x & 31;
    const int wib  = threadIdx.x >> 5;
    const int g    = lane >> 4;       // half-wave group
    const int c    = lane & 15;       // column within tile
    float* ldsw    = &lds[wib][0];

    const int wave = blockIdx.x * WAVES_PER_BLOCK + wib;   // one tile per wave
    const int bimg = wave >> 10;                           // / TILES_PER_IMG
    const int t    = wave & (TILES_PER_IMG - 1);
    const int row0 = (t >> 5) << 4;
    const int col0 = (t & 31) << 4;

    const float* inb = in + (size_t)bimg * 3 * HW;
    float* outb      = out + (size_t)bimg * 3 * HW;

    // ---- load RGB (non-temporal) in B layout, color transform, clip ------
    float Yc[8], Cb[8], Cr[8];
#pragma unroll
    for (int kk = 0; kk < 4; ++kk)
#pragma unroll
        for (int j = 0; j < 2; ++j) {
            const int p   = kk * 2 + j;
            const int row = row0 + 4 * kk + 2 * g + j;
            const int off = row * IMG_W + col0 + c;
            float r  = 255.0f * __builtin_nontemporal_load(inb + off);
            float gg = 255.0f * __builtin_nontemporal_load(inb + HW + off);
            float bb = 255.0f * __builtin_nontemporal_load(inb + 2 * HW + off);
            Yc[p] = fminf(fmaxf(0.299f * r + 0.587f * gg + 0.114f * bb, 0.f), 255.f);
            Cb[p] = fminf(fmaxf(-0.168735892f * r - 0.331264108f * gg + 0.5f * bb + 128.f, 0.f), 255.f);
            Cr[p] = fminf(fmaxf(0.5f * r - 0.418687589f * gg - 0.081312411f * bb + 128.f, 0.f), 255.f);
        }

    // ---- 2x2 chroma mean (rows: adjacent j regs; cols: lane ^ 1) ---------
#pragma unroll
    for (int kk = 0; kk < 4; ++kk) {
        float sb = Cb[2 * kk] + Cb[2 * kk + 1];
        float mb = 0.25f * (sb + __shfl_xor(sb, 1, 32));
        Cb[2 * kk] = Cb[2 * kk + 1] = mb;
        float sr = Cr[2 * kk] + Cr[2 * kk + 1];
        float mr = 0.25f * (sr + __shfl_xor(sr, 1, 32));
        Cr[2 * kk] = Cr[2 * kk + 1] = mr;
    }
#pragma unroll
    for (int p = 0; p < 8; ++p) {
        Yc[p] -= 128.f; Cb[p] -= 128.f; Cr[p] -= 128.f;
    }

    // ---- block-diagonal 16x16 DCT matrix in WMMA A layout ----------------
    float aM[8], aMt[8];
#pragma unroll
    for (int kk = 0; kk < 4; ++kk)
#pragma unroll
        for (int j = 0; j < 2; ++j) {
            const int p = kk * 2 + j;
            const int k = 4 * kk + 2 * g + j;   // A-layout K index
            const int m = c;                    // A-layout M index (lane % 16)
            const bool same = (m >> 3) == (k >> 3);
            aM[p]  = same ? dct[(m & 7) * 8 + (k & 7)] : 0.0f;   // M[m][k]
            aMt[p] = same ? dct[(k & 7) * 8 + (m & 7)] : 0.0f;   // Mt[m][k]
        }

    // ---- per-channel DCT -> quant -> IDCT --------------------------------
    float z0[8], z1[8], z2[8];
    dct_pipeline(Yc, aM, aMt, qtab + 0,   g, c, ldsw, z0);
    dct_pipeline(Cb, aM, aMt, qtab + 64,  g, c, ldsw, z1);
    dct_pipeline(Cr, aM, aMt, qtab + 128, g, c, ldsw, z2);

    // ---- inverse color transform, round, store (non-temporal, coalesced) -
#pragma unroll
    for (int r = 0; r < 8; ++r) {
        float y  = z0[r] + 128.f;  // ycbcr2 - off: Y keeps +128
        float cb = z1[r];          // +128 - 128
        float cr = z2[r];
        float R = y + 1.402f * cr;
        float G = y - 0.344136286f * cb - 0.714136286f * cr;
        float Bv = y + 1.772f * cb;
        R  = rintf(fminf(fmaxf(R, 0.f), 255.f)) * (1.0f / 255.0f);
        G  = rintf(fminf(fmaxf(G, 0.f), 255.f)) * (1.0f / 255.0f);
        Bv = rintf(fminf(fmaxf(Bv, 0.f), 255.f)) * (1.0f / 255.0f);
        const int row = row0 + g * 8 + r;
        const int off = row * IMG_W + col0 + c;
        __builtin_nontemporal_store(R,  outb + off);
        __builtin_nontemporal_store(G,  outb + HW + off);
        __builtin_nontemporal_store(Bv, outb + 2 * HW + off);
    }
}

extern "C" void kernel_launch(void* const* d_in, const int* in_sizes, int n_in,
                              void* d_out, int out_size, void* d_ws, size_t ws_size,
                              hipStream_t stream) {
    (void)in_sizes; (void)n_in; (void)out_size; (void)d_ws; (void)ws_size;
    const float* x   = (const float*)d_in[0];   // (32,3,512,512)
    const float* q   = (const float*)d_in[1];   // (3,8,8)
    const float* dmx = (const float*)d_in[2];   // (8,8)
    float* out       = (float*)d_out;
    jpeg_layer_kernel<<<BLOCKS, WAVES_PER_BLOCK * 32, 0, stream>>>(x, q, dmx, out);
}